// DiGCN_68659347194091
// MI455X (gfx1250) — compile-verified
//
#include <hip/hip_runtime.h>
#include <math.h>

typedef __attribute__((ext_vector_type(2))) float v2f;
typedef __attribute__((ext_vector_type(8))) float v8f;

#define DFEAT 128

// ---------------------------------------------------------------------------
// init: deg = 1.0 (self-loop pre-counted), zero h1-accumulator and d_out
// ---------------------------------------------------------------------------
__global__ void tgcn_init(float* __restrict__ deg, float* __restrict__ acc1,
                          float* __restrict__ acc2, int n_nodes, size_t nd) {
  size_t i = (size_t)blockIdx.x * blockDim.x + threadIdx.x;
  if (i < nd) {
    acc1[i] = 0.0f;
    acc2[i] = 0.0f;
  }
  if (i < (size_t)n_nodes) deg[i] = 1.0f;
}

// ---------------------------------------------------------------------------
// degree: deg[dst[e]] += 1
// ---------------------------------------------------------------------------
__global__ void tgcn_degree(const int* __restrict__ dst, float* __restrict__ deg,
                            int n_edges) {
  int e = blockIdx.x * blockDim.x + threadIdx.x;
  if (e < n_edges) atomicAdd(&deg[dst[e]], 1.0f);
}

// ---------------------------------------------------------------------------
// dinv = rsqrt(deg) in place (deg >= 1 always, self-loops counted)
// ---------------------------------------------------------------------------
__global__ void tgcn_rsqrt(float* __restrict__ deg, int n_nodes) {
  int i = blockIdx.x * blockDim.x + threadIdx.x;
  if (i < n_nodes) deg[i] = rsqrtf(deg[i]);
}

// ---------------------------------------------------------------------------
// WMMA fp32 GEMM:  Y[N,128] = A[N,128] @ W[128,128]
// Optionally fuses TimeEncode into A load: A = X + cos(t*freq + phase).
// One block = 256 threads = 8 waves; block owns a 16-row M tile; wave w owns
// output columns [16w, 16w+16). K=128 via 32 x V_WMMA_F32_16X16X4_F32.
// ---------------------------------------------------------------------------
__global__ void __launch_bounds__(256) tgcn_gemm_wmma(
    const float* __restrict__ X, const float* __restrict__ W,
    const float* __restrict__ t, const float* __restrict__ freq,
    const float* __restrict__ phase, int use_te,
    float* __restrict__ Y, int n_nodes) {
  __shared__ float sA[16 * DFEAT];  // 8 KB A tile (with fused time-encode)

  const int tid = threadIdx.x;
  const int mbase = blockIdx.x * 16;

  // cooperative A-tile load: 2048 elements / 256 threads = 8 each
  for (int i = tid; i < 16 * DFEAT; i += 256) {
    int r = i >> 7;
    int c = i & (DFEAT - 1);
    int row = mbase + r;
    float v = 0.0f;
    if (row < n_nodes) {
      v = X[(size_t)row * DFEAT + c];
      if (use_te) v += cosf(t[row] * freq[c] + phase[c]);
    }
    sA[i] = v;
  }
  __syncthreads();

  const int wave = tid >> 5;
  const int lane = tid & 31;
  const int half = lane >> 4;        // 0: K pair {0,1}, 1: K pair {2,3}
  const int arow = lane & 15;        // A-matrix row within tile
  const int ncol = wave * 16 + (lane & 15);

  v8f acc = {};
  for (int k = 0; k < DFEAT; k += 4) {
    const int ka = k + 2 * half;
    v2f a, b;
    a.x = sA[arow * DFEAT + ka];
    a.y = sA[arow * DFEAT + ka + 1];
    b.x = W[(size_t)ka * DFEAT + ncol];
    b.y = W[(size_t)(ka + 1) * DFEAT + ncol];
    // D = A(16x4,f32) * B(4x16,f32) + C(16x16,f32)
    acc = __builtin_amdgcn_wmma_f32_16x16x4_f32(
        /*neg_a=*/false, a, /*neg_b=*/false, b,
        /*c_mod=*/(short)0, acc, /*reuse_a=*/false, /*reuse_b=*/false);
  }

  // C/D layout: VGPR r -> row (r + 8*half), col (lane&15)
#pragma unroll
  for (int r = 0; r < 8; ++r) {
    int orow = mbase + r + 8 * half;
    if (orow < n_nodes) Y[(size_t)orow * DFEAT + ncol] = acc[r];
  }
}

// ---------------------------------------------------------------------------
// edge scatter: one wave32 per edge; lane carries float4 (128 feats / 32 lanes)
// out[dst,:] += xw[src,:] * dinv[src]*dinv[dst]   (f32 global atomics -> L2)
// ---------------------------------------------------------------------------
__global__ void __launch_bounds__(256) tgcn_scatter(
    const float* __restrict__ xw, const int* __restrict__ src,
    const int* __restrict__ dst, const float* __restrict__ dinv,
    float* __restrict__ out, int n_edges) {
  const int lane = threadIdx.x & 31;
  const int edge = blockIdx.x * (blockDim.x >> 5) + (threadIdx.x >> 5);
  if (edge >= n_edges) return;

  const int s = src[edge];          // wave-uniform -> scalar loads
  const int d = dst[edge];
  const float norm = dinv[s] * dinv[d];

  const float4 v = ((const float4*)(xw + (size_t)s * DFEAT))[lane];
  float* o = out + (size_t)d * DFEAT + lane * 4;
  atomicAdd(o + 0, v.x * norm);
  atomicAdd(o + 1, v.y * norm);
  atomicAdd(o + 2, v.z * norm);
  atomicAdd(o + 3, v.w * norm);
}

// ---------------------------------------------------------------------------
// self-loop + bias: out[i,c] += xw[i,c]*dinv[i]^2 + b[c]
// Runs after the edge-scatter kernel (stream-serialized) -> plain RMW, no
// atomics; each (i,c) touched exactly once.
// ---------------------------------------------------------------------------
__global__ void tgcn_selfloop_bias(const float* __restrict__ xw,
                                   const float* __restrict__ dinv,
                                   const float* __restrict__ bias,
                                   float* __restrict__ out, size_t nd) {
  size_t i = (size_t)blockIdx.x * blockDim.x + threadIdx.x;
  if (i >= nd) return;
  int node = (int)(i >> 7);
  int c = (int)(i & (DFEAT - 1));
  float dv = dinv[node];
  out[i] += xw[i] * dv * dv + bias[c];
}

// ---------------------------------------------------------------------------
extern "C" void kernel_launch(void* const* d_in, const int* in_sizes, int n_in,
                              void* d_out, int out_size, void* d_ws, size_t ws_size,
                              hipStream_t stream) {
  const float* x     = (const float*)d_in[0];
  const int*   ei    = (const int*)d_in[1];   // [2, E] row-major
  const float* t     = (const float*)d_in[2];
  const float* W1    = (const float*)d_in[3];
  const float* b1    = (const float*)d_in[4];
  const float* W2    = (const float*)d_in[5];
  const float* b2    = (const float*)d_in[6];
  const float* freq  = (const float*)d_in[7];
  const float* phase = (const float*)d_in[8];

  const int n_nodes = in_sizes[2];            // t_index: [N]
  const int n_edges = in_sizes[1] / 2;
  const size_t nd = (size_t)n_nodes * DFEAT;

  const int* src = ei;                        // edge_index[0]
  const int* dst = ei + n_edges;              // edge_index[1]

  float* out = (float*)d_out;

  // workspace layout: [ dinv: N floats | bufA: N*D | bufB: N*D ]
  float* dinv = (float*)d_ws;
  size_t noff = ((size_t)n_nodes + 255) & ~(size_t)255;
  float* bufA = dinv + noff;                  // xw1, later xw2
  float* bufB = bufA + nd;                    // h1 accumulator

  const int T = 256;
  const int g_nd   = (int)((nd + T - 1) / T);
  const int g_edge = (n_edges + T - 1) / T;
  const int g_node = (n_nodes + T - 1) / T;
  const int g_gemm = (n_nodes + 15) / 16;
  const int g_scat = (n_edges + 7) / 8;       // 8 waves per 256-thread block

  // deg=1 (self loop), zero accumulators (d_out is poisoned each timing run)
  tgcn_init<<<g_nd, T, 0, stream>>>(dinv, bufB, out, n_nodes, nd);
  // degree + symmetric normalization
  tgcn_degree<<<g_edge, T, 0, stream>>>(dst, dinv, n_edges);
  tgcn_rsqrt<<<g_node, T, 0, stream>>>(dinv, n_nodes);

  // layer 1: xw1 = (x + cos(t*freq + phase)) @ W1
  tgcn_gemm_wmma<<<g_gemm, T, 0, stream>>>(x, W1, t, freq, phase, 1, bufA, n_nodes);
  tgcn_scatter<<<g_scat, T, 0, stream>>>(bufA, src, dst, dinv, bufB, n_edges);
  tgcn_selfloop_bias<<<g_nd, T, 0, stream>>>(bufA, dinv, b1, bufB, nd);

  // layer 2: xw2 = h1 @ W2 ; h2 -> d_out
  tgcn_gemm_wmma<<<g_gemm, T, 0, stream>>>(bufB, W2, t, freq, phase, 0, bufA, n_nodes);
  tgcn_scatter<<<g_scat, T, 0, stream>>>(bufA, src, dst, dinv, out, n_edges);
  tgcn_selfloop_bias<<<g_nd, T, 0, stream>>>(bufA, dinv, b2, out, nd);
}